// ModelNew_4810363372160
// MI455X (gfx1250) — compile-verified
//
#include <hip/hip_runtime.h>

typedef __attribute__((ext_vector_type(2))) float v2f;
typedef __attribute__((ext_vector_type(4))) float v4f;
typedef __attribute__((ext_vector_type(8))) float v8f;
typedef __attribute__((ext_vector_type(4))) unsigned int u32x4;
typedef __attribute__((ext_vector_type(8))) int i32x8;
typedef __attribute__((ext_vector_type(4))) int i32x4;

#define NN 4096
#define BM 128
#define BN 128
#define BK 16
#define LDA 20    // A tile LDS stride (16 data + 4 pad DWORDs) -> conflict-free frags
#define LDBT 20   // transposed-B tile LDS stride

#if defined(__has_builtin)
#if __has_builtin(__builtin_amdgcn_tensor_load_to_lds) && \
    __has_builtin(__builtin_amdgcn_s_wait_tensorcnt)
#define USE_TDM 1
#endif
#endif

#define WMMA_F32(a, b, c) \
  __builtin_amdgcn_wmma_f32_16x16x4_f32(false, (a), false, (b), (short)0, (c), false, false)

// ---- B tile 16 x 128: global float4 load (registers), later transpose-scatter
//      into Bt[n][k] as 2x ds_store_2addr_b32.
__device__ __forceinline__ v4f load_B(const float* __restrict__ B, int j0,
                                      int kc, int tid) {
  const int brow = tid >> 5;         // 0..15  (k within chunk)
  const int bcol = (tid & 31) * 4;   // 0..124 (n within tile)
  return *(const v4f*)(B + (size_t)(kc + brow) * NN + (size_t)(j0 + bcol));
}

__device__ __forceinline__ void store_B(v4f bv, float* BtBuf, int tid) {
  const int brow = tid >> 5;
  const int bcol = (tid & 31) * 4;
#pragma unroll
  for (int e = 0; e < 4; ++e)
    BtBuf[(bcol + e) * LDBT + brow] = bv[e];
}

#if USE_TDM
// ---- A tile 128 x 16 via Tensor Data Mover: one DMA per chunk, issued by
//      wave 0; descriptor padding (16 DW data + 4 DW pad) produces stride LDA.
__device__ __forceinline__ void issue_A(const float* __restrict__ A, int i0,
                                        int kc, float* AsBuf, int wave) {
  if (wave == 0) {
    const unsigned long long ga =
        (unsigned long long)(size_t)(A + (size_t)i0 * NN + kc);
    const unsigned int lds = (unsigned int)(size_t)(void*)AsBuf;
    u32x4 g0;
    g0.x = 1u;                                   // count=1, user descriptor
    g0.y = lds;                                  // lds_addr
    g0.z = (unsigned int)ga;                     // global_addr[31:0]
    g0.w = (unsigned int)((ga >> 32) & 0x1FFFFFFull) | (2u << 30);  // addr[56:32] | type=2
    i32x8 g1;
    g1[0] = (int)((2u << 16)      // data_size = 4B
                  | (1u << 20)    // pad_enable
                  | (3u << 22)    // pad_interval: 16 DWORDs
                  | (3u << 25));  // pad_amount: 4 DWORDs -> LDS row stride 20
    g1[1] = (int)(((unsigned)NN & 0xFFFFu) << 16);  // tensor_dim0[15:0]
    g1[2] = (int)(((unsigned)NN & 0xFFFFu) << 16);  // dim0[31:16]=0 | tensor_dim1[15:0]
    g1[3] = (int)(16u << 16);                       // dim1[31:16]=0 | tile_dim0 = 16
    g1[4] = 128;                                    // tile_dim1 = 128 (tile_dim2 = 0)
    g1[5] = NN;                                     // tensor_dim0_stride = 4096
    g1[6] = 0;
    g1[7] = 0;
    i32x4 z4 = {};
    i32x8 z8 = {};
    __builtin_amdgcn_tensor_load_to_lds(g0, g1, z4, z4, z8, 0);
  }
}
#else
// ---- A tile 128 x 16, cooperative float4 staging (fallback, no TDM builtin)
__device__ __forceinline__ v4f load_A(const float* __restrict__ A, int i0,
                                      int kc, int tid) {
  const int arow = tid >> 2;        // 0..127
  const int aq = (tid & 3) * 4;     // 0..12
  return *(const v4f*)(A + (size_t)(i0 + arow) * NN + kc + aq);
}
__device__ __forceinline__ void store_A(v4f av, float* AsBuf, int tid) {
  const int arow = tid >> 2;
  const int aq = (tid & 3) * 4;
  *(v4f*)&AsBuf[arow * LDA + aq] = av;
}
#endif

__global__ __launch_bounds__(512) void triu_gemm_f32_wmma(
    const float* __restrict__ A, const float* __restrict__ B,
    float* __restrict__ C) {
  const int j0 = blockIdx.x * BN;  // tile column start
  const int i0 = blockIdx.y * BM;  // tile row start
  const int tid = threadIdx.x;

  // ---- Tiles entirely below the diagonal: write zeros, no compute.
  if (i0 > j0) {
    const int row = tid >> 2;      // 0..127
    const int q = (tid & 3) * 4;   // 0,4,8,12
    float* out = C + (size_t)(i0 + row) * NN + j0 + q;
    v4f z = {};
#pragma unroll
    for (int jj = 0; jj < 8; ++jj)
      *(v4f*)(out + jj * 16) = z;
    return;
  }

  __shared__ float As[2][BM * LDA];   // 128 x 16 (+pad), double buffered
  __shared__ float Bt[2][BN * LDBT];  // 128 x 16 transposed B (+pad), double buffered

  const int lane = tid & 31;
  const int wave = tid >> 5;          // 0..15 (16 wave32s)
  const int wm32 = (wave & 3) * 32;   // 32-row band
  const int wn32 = (wave >> 2) * 32;  // 32-col band

  const int mrow = lane & 15;         // M (A frag) / N (B frag) within 16
  const int khalf = (lane >> 4) * 2;  // lanes 0-15 -> K0,K1 ; 16-31 -> K2,K3

  v8f c00 = {}, c01 = {}, c10 = {}, c11 = {};

  // Triangular k-range: A[i,k]==0 for k<i, B[k,j]==0 for k>j.
  const int nch = (j0 + BN - i0) >> 4;  // >= 8 chunks of BK=16

  // ---- Prologue: stage chunk 0 into buffer 0.
  {
    v4f bv = load_B(B, j0, i0, tid);
#if USE_TDM
    issue_A(A, i0, i0, &As[0][0], wave);
#else
    v4f av = load_A(A, i0, i0, tid);
    store_A(av, &As[0][0], tid);
#endif
    store_B(bv, &Bt[0][0], tid);
  }
#if USE_TDM
  __builtin_amdgcn_s_wait_tensorcnt(0);
#endif
  __syncthreads();

  for (int c = 0; c < nch; ++c) {
    const int cur = c & 1;
    const bool have_next = (c + 1 < nch);

    // Issue next chunk's data movement EARLY (latency hidden by the WMMAs):
    //  - TDM DMA for A (async, TENSORcnt)
    //  - global_load_b128 for B into registers (LOADcnt waited after compute)
    v4f bvn;
#if !USE_TDM
    v4f avn;
#endif
    if (have_next) {
      const int kn = i0 + (c + 1) * BK;
      bvn = load_B(B, j0, kn, tid);
#if USE_TDM
      issue_A(A, i0, kn, &As[1 - cur][0], wave);
#else
      avn = load_A(A, i0, kn, tid);
#endif
    }

    // ---- 16 WMMAs on the current LDS tiles
    const float* as = &As[cur][0];
    const float* bt = &Bt[cur][0];
#pragma unroll
    for (int kk = 0; kk < BK; kk += 4) {
      v2f a0 = *(const v2f*)(as + (wm32 + mrow) * LDA + kk + khalf);
      v2f a1 = *(const v2f*)(as + (wm32 + 16 + mrow) * LDA + kk + khalf);
      v2f b0 = *(const v2f*)(bt + (wn32 + mrow) * LDBT + kk + khalf);
      v2f b1 = *(const v2f*)(bt + (wn32 + 16 + mrow) * LDBT + kk + khalf);
      c00 = WMMA_F32(a0, b0, c00);
      c01 = WMMA_F32(a0, b1, c01);
      c10 = WMMA_F32(a1, b0, c10);
      c11 = WMMA_F32(a1, b1, c11);
    }

    // LDS stores for the next buffer AFTER compute (s_wait_loadcnt lands here)
    if (have_next) {
      store_B(bvn, &Bt[1 - cur][0], tid);
#if !USE_TDM
      store_A(avn, &As[1 - cur][0], tid);
#endif
    }

#if USE_TDM
    __builtin_amdgcn_s_wait_tensorcnt(0);  // next A tile landed (wave 0; no-op others)
#endif
    __syncthreads();  // drains DScnt for B stores; publishes next buffer
  }

  // ---- Store with triu mask (C/D layout: VGPR v holds M = v + 8*(lane>>4))
  const int mb = i0 + wm32 + ((lane >> 4) * 8);
  const int nb = j0 + wn32 + (lane & 15);
#pragma unroll
  for (int v = 0; v < 8; ++v) {
    const int m0 = mb + v;
    const int m1 = m0 + 16;
    C[(size_t)m0 * NN + nb]      = (m0 <= nb)      ? c00[v] : 0.0f;
    C[(size_t)m0 * NN + nb + 16] = (m0 <= nb + 16) ? c01[v] : 0.0f;
    C[(size_t)m1 * NN + nb]      = (m1 <= nb)      ? c10[v] : 0.0f;
    C[(size_t)m1 * NN + nb + 16] = (m1 <= nb + 16) ? c11[v] : 0.0f;
  }
}

extern "C" void kernel_launch(void* const* d_in, const int* in_sizes, int n_in,
                              void* d_out, int out_size, void* d_ws, size_t ws_size,
                              hipStream_t stream) {
  const float* A = (const float*)d_in[0];
  const float* B = (const float*)d_in[1];
  float* C = (float*)d_out;
  dim3 grid(NN / BN, NN / BM);
  triu_gemm_f32_wmma<<<grid, dim3(512), 0, stream>>>(A, B, C);
}